// BlockAttentionResidual_41180146434486
// MI455X (gfx1250) — compile-verified
//
#include <hip/hip_runtime.h>
#include <stdint.h>

// Problem constants from the reference: B=4, T=2048, D=2048, K=8 -> S=9 sources.
#define S 9
#define DDIM 2048
#define NTHREADS 256
#define CHUNKS (DDIM / 4 / NTHREADS)   // float4 chunks per thread = 2
#define EPS 1e-8f

typedef float v4f __attribute__((ext_vector_type(4)));

// CDNA5 async global -> LDS copy (ASYNCcnt-tracked, no VGPR staging).
// Source data is stream-once -> non-temporal hint keeps L2 free for the
// heavily reused query/key_weight vectors.
// %0 = 32-bit LDS byte offset (low 32 bits of a generic LDS pointer == LDS
//      offset per the aperture rule, ISA 10.2), %1 = 64-bit global address.
__device__ __forceinline__ void async_load_b128_nt(uint32_t lds_off, const void* gaddr) {
    asm volatile("global_load_async_to_lds_b128 %0, %1, off th:TH_LOAD_NT"
                 :: "v"(lds_off), "v"((uint64_t)(uintptr_t)gaddr)
                 : "memory");
}

__device__ __forceinline__ void wait_async_zero() {
#if __has_builtin(__builtin_amdgcn_s_wait_asynccnt)
    __builtin_amdgcn_s_wait_asynccnt(0);
#else
    asm volatile("s_wait_asynccnt 0" ::: "memory");
#endif
}

__global__ __launch_bounds__(NTHREADS)
void BlockAttentionResidual_41180146434486_kernel(
    const float* __restrict__ embedding,   // [rows, D]
    const float* __restrict__ blocks,      // [K, rows, D]
    const float* __restrict__ query,       // [D]
    const float* __restrict__ key_weight,  // [D]
    float* __restrict__ out_routed,        // [rows, D]
    float* __restrict__ out_weights,       // [rows, S]
    float* __restrict__ out_entropy,       // [rows]
    int rows)
{
    __shared__ __align__(16) float lds_src[S * DDIM];   // 73728 B
    __shared__ float lds_red[2 * S];                    // [0..8]=sumsq, [9..17]=dot

    const int row = blockIdx.x;
    const int tid = threadIdx.x;

    // LDS byte offset of the staging buffer (generic-LDS pointer low bits).
    const uint32_t lds_src_off = (uint32_t)(uintptr_t)(&lds_src[0]);

    // ---- Stage the full 9 x 2048 fp32 row into LDS with async NT copies ----
    #pragma unroll
    for (int s = 0; s < S; ++s) {
        const float* srcp = (s == 0)
            ? (embedding + (size_t)row * DDIM)
            : (blocks + ((size_t)(s - 1) * rows + row) * DDIM);
        #pragma unroll
        for (int c = 0; c < CHUNKS; ++c) {
            const int j = tid + c * NTHREADS;  // float4 index within the D row
            const uint32_t loff =
                lds_src_off + (uint32_t)((s * DDIM + j * 4) * sizeof(float));
            async_load_b128_nt(loff, srcp + (size_t)j * 4);
        }
    }

    if (tid < 2 * S) lds_red[tid] = 0.0f;

    wait_async_zero();     // my wave's async copies landed in LDS
    __syncthreads();       // everyone's landed + lds_red zeroed

    // ---- Pass A: per-source sum(src^2) and dot(src, q*w) ----
    float acc_sq[S], acc_dot[S];
    #pragma unroll
    for (int s = 0; s < S; ++s) { acc_sq[s] = 0.0f; acc_dot[s] = 0.0f; }

    const float4* lsrc4 = (const float4*)lds_src;
    #pragma unroll
    for (int c = 0; c < CHUNKS; ++c) {
        const int j = tid + c * NTHREADS;
        const float4 q4 = ((const float4*)query)[j];       // L2-hot (RT policy)
        const float4 w4 = ((const float4*)key_weight)[j];  // L2-hot (RT policy)
        const float qwx = q4.x * w4.x, qwy = q4.y * w4.y;
        const float qwz = q4.z * w4.z, qww = q4.w * w4.w;
        #pragma unroll
        for (int s = 0; s < S; ++s) {
            const float4 v = lsrc4[s * (DDIM / 4) + j];
            acc_sq[s]  += v.x * v.x + v.y * v.y + v.z * v.z + v.w * v.w;
            acc_dot[s] += v.x * qwx + v.y * qwy + v.z * qwz + v.w * qww;
        }
    }

    // wave32 butterfly reduction, then one ds_add_f32 per wave per accumulator
    #pragma unroll
    for (int s = 0; s < S; ++s) {
        float a = acc_sq[s], b = acc_dot[s];
        #pragma unroll
        for (int off = 16; off > 0; off >>= 1) {
            a += __shfl_xor(a, off, 32);
            b += __shfl_xor(b, off, 32);
        }
        if ((tid & 31) == 0) {
            atomicAdd(&lds_red[s],     a);
            atomicAdd(&lds_red[S + s], b);
        }
    }
    __syncthreads();

    // ---- Softmax over the 9 sources (redundant per thread; trivial work) ----
    float logit[S];
    #pragma unroll
    for (int s = 0; s < S; ++s) {
        const float rms = sqrtf(lds_red[s] * (1.0f / (float)DDIM) + EPS);
        logit[s] = lds_red[S + s] / rms;
    }
    float mx = logit[0];
    #pragma unroll
    for (int s = 1; s < S; ++s) mx = fmaxf(mx, logit[s]);
    float wv[S];
    float wsum = 0.0f;
    #pragma unroll
    for (int s = 0; s < S; ++s) { wv[s] = expf(logit[s] - mx); wsum += wv[s]; }
    const float winv = 1.0f / wsum;
    #pragma unroll
    for (int s = 0; s < S; ++s) wv[s] *= winv;

    // ---- Pass B: weighted combine straight out of LDS; NT streaming stores ----
    v4f* orow4 = (v4f*)(out_routed + (size_t)row * DDIM);
    #pragma unroll
    for (int c = 0; c < CHUNKS; ++c) {
        const int j = tid + c * NTHREADS;
        float ax = 0.f, ay = 0.f, az = 0.f, aw = 0.f;
        #pragma unroll
        for (int s = 0; s < S; ++s) {
            const float4 v = lsrc4[s * (DDIM / 4) + j];
            ax += wv[s] * v.x; ay += wv[s] * v.y;
            az += wv[s] * v.z; aw += wv[s] * v.w;
        }
        v4f res = {ax, ay, az, aw};
        __builtin_nontemporal_store(res, orow4 + j);
    }

    // ---- weights + entropy ----
    if (tid == 0) {
        float ent = 0.0f;
        #pragma unroll
        for (int s = 0; s < S; ++s) {
            out_weights[(size_t)row * S + s] = wv[s];
            const float sw = fmaxf(wv[s], 1e-12f);
            ent -= sw * logf(sw);
        }
        out_entropy[row] = ent;
    }
}

extern "C" void kernel_launch(void* const* d_in, const int* in_sizes, int n_in,
                              void* d_out, int out_size, void* d_ws, size_t ws_size,
                              hipStream_t stream) {
    const float* embedding  = (const float*)d_in[0];  // [B,T,D]
    const float* blocks     = (const float*)d_in[1];  // [K,B,T,D]
    const float* query      = (const float*)d_in[2];  // [D]
    const float* key_weight = (const float*)d_in[3];  // [D]

    const int D    = in_sizes[2];            // 2048
    const int rows = in_sizes[0] / D;        // B*T = 8192
    (void)n_in; (void)d_ws; (void)ws_size; (void)out_size; (void)D;

    float* out_routed  = (float*)d_out;                          // rows*D
    float* out_weights = out_routed + (size_t)rows * DDIM;       // rows*S
    float* out_entropy = out_weights + (size_t)rows * S;         // rows

    BlockAttentionResidual_41180146434486_kernel<<<rows, NTHREADS, 0, stream>>>(
        embedding, blocks, query, key_weight,
        out_routed, out_weights, out_entropy, rows);
}